// SparseGraphAttentionLayer_61495341744392
// MI455X (gfx1250) — compile-verified
//
#include <hip/hip_runtime.h>
#include <hip/hip_bf16.h>

#define N_NODES  50000
#define IN_FEAT  512
#define OUT_FEAT 256
#define N_EDGES  1600000
#define ALPHA    0.2f

typedef __attribute__((ext_vector_type(2))) float v2f;
typedef __attribute__((ext_vector_type(8))) float v8f;

// ---------------------------------------------------------------------------
// Kernel 1: fold W into the two attention vectors.
//   wfold[k][0] = sum_n W[k][n] * attn[n]         (w_src)
//   wfold[k][1] = sum_n W[k][n] * attn[256 + n]   (w_tgt)
// ---------------------------------------------------------------------------
__global__ void gat_fold_kernel(const float* __restrict__ W,
                                const float* __restrict__ attn,
                                float* __restrict__ wfold) {
    int k = blockIdx.x * blockDim.x + threadIdx.x;
    if (k >= IN_FEAT) return;
    const float* row = W + (size_t)k * OUT_FEAT;
    float s0 = 0.0f, s1 = 0.0f;
    #pragma unroll 4
    for (int n = 0; n < OUT_FEAT; ++n) {
        float w = row[n];
        s0 += w * attn[n];
        s1 += w * attn[OUT_FEAT + n];
    }
    wfold[2 * k + 0] = s0;
    wfold[2 * k + 1] = s1;
}

// ---------------------------------------------------------------------------
// Kernel 2: s[n][0..1] = h[n][:] . wfold[:][0..1] via V_WMMA_F32_16X16X4_F32.
// 256 threads = 8 wave32; each wave owns a 16-row A tile; block = 128 rows.
// h staged through LDS in 128x64 chunks, row pitch 68 floats (bank-safe).
// ---------------------------------------------------------------------------
#define BROWS   128
#define KCHUNK  64
#define LPITCH  68   // 68 % 64 == 4 -> rows 0..15 hit distinct banks at same k

__global__ __launch_bounds__(256) void gat_score_kernel(
        const float* __restrict__ h,
        const float* __restrict__ wfold,
        float* __restrict__ s) {
    __shared__ __align__(16) float lh[BROWS * LPITCH];
    __shared__ float lw[IN_FEAT * 2];

    const int tid  = threadIdx.x;
    const int lane = tid & 31;
    const int wave = tid >> 5;
    const int half = lane >> 4;   // 0: lanes 0-15, 1: lanes 16-31
    const int lmod = lane & 15;
    const int row_block = blockIdx.x * BROWS;

    // stage folded vectors (4 KB) once
    for (int i = tid; i < IN_FEAT * 2; i += 256) lw[i] = wfold[i];

    const int arow = wave * 16 + lmod;      // local row this lane feeds into A
    v8f c = {};                             // 16x16 f32 accumulator tile

    for (int kc = 0; kc < IN_FEAT; kc += KCHUNK) {
        // ---- cooperative coalesced load: 128 rows x 64 cols of h ----------
        #pragma unroll
        for (int i = 0; i < (BROWS * KCHUNK / 4) / 256; ++i) {
            int idx = tid + 256 * i;        // float4 index
            int r   = idx >> 4;             // 16 float4 per row
            int c4  = idx & 15;
            int gr  = row_block + r;
            float4 v = make_float4(0.f, 0.f, 0.f, 0.f);
            if (gr < N_NODES)
                v = *(const float4*)(h + (size_t)gr * IN_FEAT + kc + c4 * 4);
            *(float4*)&lh[r * LPITCH + c4 * 4] = v;
        }
        __syncthreads();

        // ---- 16 WMMA K-steps over this chunk ------------------------------
        #pragma unroll
        for (int ks = 0; ks < KCHUNK; ks += 4) {
            const int kg = kc + ks;
            // A frag (16x4 f32): reg j, half hf -> K = ks + j + 2*hf, M = lmod
            v2f a;
            a.x = lh[arow * LPITCH + ks + 2 * half + 0];
            a.y = lh[arow * LPITCH + ks + 2 * half + 1];
            // B frag (4x16 f32): reg j -> K = kg + j + 2*half, N = lmod.
            // Only N==0 (w_src) and N==1 (w_tgt) are nonzero.
            float b0 = lw[(kg + 2 * half + 0) * 2 + (lmod & 1)];
            float b1 = lw[(kg + 2 * half + 1) * 2 + (lmod & 1)];
            v2f b;
            b.x = (lmod < 2) ? b0 : 0.0f;
            b.y = (lmod < 2) ? b1 : 0.0f;
            c = __builtin_amdgcn_wmma_f32_16x16x4_f32(
                    false, a, false, b, (short)0, c, false, false);
        }
        __syncthreads();
    }

    // C layout: reg j holds row M=j (lanes 0-15) / M=j+8 (lanes 16-31), N=lmod
    #pragma unroll
    for (int j = 0; j < 8; ++j) {
        int gr = row_block + wave * 16 + j + 8 * half;
        if (lmod < 2 && gr < N_NODES)
            s[(size_t)gr * 2 + lmod] = c[j];
    }
}

// ---------------------------------------------------------------------------
// Kernel 3: per-edge gather + leaky_relu.
// ---------------------------------------------------------------------------
__global__ __launch_bounds__(256) void gat_edge_kernel(
        const long long* __restrict__ edge_list,
        const float* __restrict__ s,
        float* __restrict__ out) {
    int i = blockIdx.x * blockDim.x + threadIdx.x;
    if (i >= N_EDGES) return;
    long long src = edge_list[i];
    long long tgt = edge_list[(size_t)N_EDGES + i];
    float e = s[(size_t)src * 2 + 0] + s[(size_t)tgt * 2 + 1];
    out[i] = (e > 0.0f) ? e : ALPHA * e;
}

// ---------------------------------------------------------------------------
extern "C" void kernel_launch(void* const* d_in, const int* in_sizes, int n_in,
                              void* d_out, int out_size, void* d_ws, size_t ws_size,
                              hipStream_t stream) {
    const float*     h    = (const float*)d_in[0];
    const long long* el   = (const long long*)d_in[1];
    const float*     W    = (const float*)d_in[2];
    const float*     attn = (const float*)d_in[3];
    float* out = (float*)d_out;

    float* wfold = (float*)d_ws;                    // 512*2 floats
    float* s     = (float*)d_ws + IN_FEAT * 2;      // 50000*2 floats

    gat_fold_kernel<<<(IN_FEAT + 255) / 256, 256, 0, stream>>>(W, attn, wfold);

    int score_blocks = (N_NODES + BROWS - 1) / BROWS;
    gat_score_kernel<<<score_blocks, 256, 0, stream>>>(h, wfold, s);

    gat_edge_kernel<<<(N_EDGES + 255) / 256, 256, 0, stream>>>(el, s, out);
}